// DCGRUcell_18141941858365
// MI455X (gfx1250) — compile-verified
//
#include <hip/hip_runtime.h>
#include <cstddef>

typedef __attribute__((ext_vector_type(2))) float v2f;
typedef __attribute__((ext_vector_type(8))) float v8f;

#define NN   10000
#define BB   16
#define EE   320000
#define CIN  32
#define COUT 64
#define CC   96          // CIN + COUT
#define ROW  1536        // BB * CC
#define NB   160000      // NN * BB
#define NBCC 15360000    // NB * CC

// ---------------- utility kernels ----------------

__global__ void fill0_kernel(float4* p, int n4) {
    int i = blockIdx.x * blockDim.x + threadIdx.x;
    if (i < n4) p[i] = make_float4(0.f, 0.f, 0.f, 0.f);
}

__global__ void initneg_kernel(const float4* __restrict__ a,
                               float4* __restrict__ m3, float4* __restrict__ m4, int n4) {
    int i = blockIdx.x * blockDim.x + threadIdx.x;
    if (i >= n4) return;
    float4 v = a[i];
    float4 nv = make_float4(-v.x, -v.y, -v.z, -v.w);
    m3[i] = nv;
    m4[i] = nv;
}

// ---------------- degree / edge-weight normalization ----------------

__global__ void deg_kernel(const long long* __restrict__ src, const long long* __restrict__ dst,
                           const float* __restrict__ ew,
                           float* __restrict__ deg_f, float* __restrict__ deg_b) {
    int e = blockIdx.x * blockDim.x + threadIdx.x;
    if (e >= EE) return;
    float w = ew[e];
    unsafeAtomicAdd(&deg_f[(int)src[e]], w);
    unsafeAtomicAdd(&deg_b[(int)dst[e]], w);
}

__global__ void normw_kernel(const long long* __restrict__ src, const long long* __restrict__ dst,
                             const float* __restrict__ ew,
                             const float* __restrict__ deg_f, const float* __restrict__ deg_b,
                             float* __restrict__ wf, float* __restrict__ wb) {
    int e = blockIdx.x * blockDim.x + threadIdx.x;
    if (e >= EE) return;
    float w = ew[e];
    wf[e] = w / deg_f[(int)src[e]];
    wb[e] = w / deg_b[(int)dst[e]];
}

// ---------------- build X0 = concat(x, h) in (n, b, c) layout ----------------

__global__ void build_x0_kernel(const float* __restrict__ x, const float* __restrict__ h,
                                float* __restrict__ A) {
    int i = blockIdx.x * blockDim.x + threadIdx.x;   // i = n*1536 + b*96 + c
    if (i >= NBCC) return;
    int c  = i % CC;
    int nb = i / CC;          // n*16 + b
    int b  = nb % BB;
    int n  = nb / BB;
    float v = (c < CIN) ? x[((size_t)b * NN + n) * CIN + c]
                        : h[((size_t)b * NN + n) * COUT + (c - CIN)];
    A[i] = v;
}

// ---------------- SpMM scatter-add: out[s[e]] += scale*w[e]*in[g[e]] ----------------
// one block (384 threads) per edge; each thread moves one float4 of the 1536-float row

__global__ void spmm_kernel(const float* __restrict__ in, float* __restrict__ out,
                            const float* __restrict__ w,
                            const long long* __restrict__ g, const long long* __restrict__ s,
                            float scale) {
    int e = blockIdx.x;
    int t = threadIdx.x;                       // 0..383
    long long gi = g[e];
    long long si = s[e];
    float wv = w[e] * scale;
    const float4* sp = (const float4*)(in + (size_t)gi * ROW);
    float4 v = sp[t];
    float* dp = out + (size_t)si * ROW + (size_t)t * 4;
    unsafeAtomicAdd(dp + 0, v.x * wv);
    unsafeAtomicAdd(dp + 1, v.y * wv);
    unsafeAtomicAdd(dp + 2, v.z * wv);
    unsafeAtomicAdd(dp + 3, v.w * wv);
}

// ---------------- WMMA GEMM: Out(NB,LD) = sum_l A_l(NB,96) @ P_l(96,LD) + bias ----
// P_l[k][j] = P[(k*5+l)*LD + j].  MODE 0: sigmoid; MODE 1: tanh(tanh).
// Block computes a 64 x LD tile: LD/16 waves (one j-tile each), 4 M-tiles per wave
// (4 independent WMMA accumulator chains).  A tiles staged in LDS (padded rows,
// conflict-free ds_load_b64 fragment reads).  B read from global (L2-resident).

#define APAD 100   // padded LDS row length (bank-conflict-free for 16x4 reads)

template <int LD, int MODE>
__global__ void cheb_gemm_kernel(const float* __restrict__ A0, const float* __restrict__ A1,
                                 const float* __restrict__ A2, const float* __restrict__ A3,
                                 const float* __restrict__ A4,
                                 const float* __restrict__ P, const float* __restrict__ bias,
                                 float* __restrict__ outBuf) {
    __shared__ float tileA[64][APAD];            // 25.6 KB

    const int lane   = threadIdx.x & 31;
    const int wave   = threadIdx.x >> 5;         // 0 .. LD/16-1
    const int mBase  = blockIdx.x * 64;
    const int jBase  = wave * 16;
    const int rowIT  = lane & 15;                // row within a 16-row tile
    const int koff   = (lane >> 4) * 2;          // 0 for lanes 0-15, 2 for 16-31
    const int j      = jBase + (lane & 15);      // output column (32-bit everywhere)

    v8f acc[4];
#pragma unroll
    for (int mt = 0; mt < 4; ++mt)
        acc[mt] = (v8f){0.f, 0.f, 0.f, 0.f, 0.f, 0.f, 0.f, 0.f};

    const float* As[5] = {A0, A1, A2, A3, A4};

    for (int l = 0; l < 5; ++l) {
        const float* __restrict__ Al = As[l];
        __syncthreads();
        // cooperative load of 64 x 96 A-tile into LDS (float4 global loads)
        for (int idx = threadIdx.x; idx < 64 * 24; idx += (LD / 16) * 32) {
            int r  = idx / 24;
            int c4 = (idx % 24) * 4;
            float4 v = *(const float4*)(Al + (size_t)(mBase + r) * CC + c4);
            *(float4*)&tileA[r][c4] = v;
        }
        __syncthreads();

        const float* __restrict__ Pl = P + l * LD;   // P_l[k][j] = Pl[k*5*LD + j]
#pragma unroll 4
        for (int kb = 0; kb < CC; kb += 4) {
            const int k0 = kb + koff;
            // B fragment (4x16 fp32): lane<16 -> rows k0,k0+1; lane>=16 -> k0+2,k0+3
            v2f b;
            b.x = Pl[k0 * 5 * LD + j];
            b.y = Pl[(k0 + 1) * 5 * LD + j];
#pragma unroll
            for (int mt = 0; mt < 4; ++mt) {
                v2f a;                               // A fragment from LDS (ds_load_b64)
                a.x = tileA[mt * 16 + rowIT][k0];
                a.y = tileA[mt * 16 + rowIT][k0 + 1];
                acc[mt] = __builtin_amdgcn_wmma_f32_16x16x4_f32(
                    false, a, false, b, (short)0, acc[mt], false, false);
            }
        }
    }

    // epilogue: D layout per VGPR v: lanes 0-15 -> M=v, lanes 16-31 -> M=v+8
    const int   rOff = (lane >> 4) * 8;
    const float bj   = bias[j];
#pragma unroll
    for (int mt = 0; mt < 4; ++mt) {
#pragma unroll
        for (int v = 0; v < 8; ++v) {
            int r = mBase + mt * 16 + v + rOff;
            float val = acc[mt][v] + bj;
            if (MODE == 0) {
                val = 1.0f / (1.0f + __expf(-val));      // sigmoid
            } else {
                val = tanhf(tanhf(val));                  // double tanh
            }
            outBuf[(size_t)r * LD + j] = val;
        }
    }
}

// ---------------- build xrh: overwrite h-half of A with r*h (x-half already there) ----

__global__ void build_xrh_kernel(const float* __restrict__ RU, const float* __restrict__ h,
                                 float* __restrict__ A) {
    int i = blockIdx.x * blockDim.x + threadIdx.x;   // i over NB*COUT
    if (i >= NB * COUT) return;
    int j  = i % COUT;
    int nb = i / COUT;
    int b  = nb % BB;
    int n  = nb / BB;
    float r  = RU[(size_t)nb * 128 + j];
    float hv = h[((size_t)b * NN + n) * COUT + j];
    A[(size_t)nb * CC + CIN + j] = r * hv;
}

// ---------------- final GRU combine: out = u*h + (1-u)*c ----------------

__global__ void final_kernel(const float* __restrict__ RU, const float* __restrict__ CB,
                             const float* __restrict__ h, float* __restrict__ out) {
    int i = blockIdx.x * blockDim.x + threadIdx.x;   // i = b*N*64 + n*64 + j
    if (i >= BB * NN * COUT) return;
    int j = i % COUT;
    int n = (i / COUT) % NN;
    int b = i / (COUT * NN);
    int nb = n * BB + b;
    float u  = RU[(size_t)nb * 128 + 64 + j];
    float c  = CB[(size_t)nb * COUT + j];
    float hv = h[i];
    out[i] = u * hv + (1.0f - u) * c;
}

// ---------------- host side ----------------

extern "C" void kernel_launch(void* const* d_in, const int* in_sizes, int n_in,
                              void* d_out, int out_size, void* d_ws, size_t ws_size,
                              hipStream_t stream) {
    const float*     x   = (const float*)d_in[0];
    const float*     h   = (const float*)d_in[1];
    const long long* ei  = (const long long*)d_in[2];
    const float*     ew  = (const float*)d_in[3];
    const float*     ruP = (const float*)d_in[4];
    const float*     ruB = (const float*)d_in[5];
    const float*     cP  = (const float*)d_in[6];
    const float*     cB  = (const float*)d_in[7];
    float* out = (float*)d_out;
    float* ws  = (float*)d_ws;

    const long long* src = ei;
    const long long* dst = ei + EE;

    // workspace layout (floats)
    float* deg_f = ws;                        // NN
    float* deg_b = deg_f + NN;                // NN
    float* wf    = deg_b + NN;                // EE
    float* wb    = wf + EE;                   // EE
    float* A0    = wb + EE;                   // NBCC  (xt0, later xrh)
    float* M1    = A0 + (size_t)NBCC;         // x1f
    float* M2    = M1 + (size_t)NBCC;         // x1b
    float* M3    = M2 + (size_t)NBCC;         // x2f
    float* M4    = M3 + (size_t)NBCC;         // x2b
    float* RU    = M4 + (size_t)NBCC;         // NB*128
    float* CBuf  = RU + (size_t)NB * 128;     // NB*64

    const int TB = 256;
    const int edgeBlocks = (EE + TB - 1) / TB;

    // 0) degrees + normalized weights
    fill0_kernel<<<(2 * NN / 4 + TB - 1) / TB, TB, 0, stream>>>((float4*)deg_f, 2 * NN / 4);
    deg_kernel<<<edgeBlocks, TB, 0, stream>>>(src, dst, ew, deg_f, deg_b);
    normw_kernel<<<edgeBlocks, TB, 0, stream>>>(src, dst, ew, deg_f, deg_b, wf, wb);

    // 1) X0 = concat(x, h) in (n, b, c) layout
    build_x0_kernel<<<(NBCC + TB - 1) / TB, TB, 0, stream>>>(x, h, A0);

    // ---------- Chebyshev pass 1 (r,u gates) ----------
    fill0_kernel<<<(2 * NBCC / 4 + TB - 1) / TB, TB, 0, stream>>>((float4*)M1, 2 * NBCC / 4);
    spmm_kernel<<<EE, 384, 0, stream>>>(A0, M1, wf, src, dst, 1.0f);   // x1f
    spmm_kernel<<<EE, 384, 0, stream>>>(A0, M2, wb, dst, src, 1.0f);   // x1b
    initneg_kernel<<<(NBCC / 4 + TB - 1) / TB, TB, 0, stream>>>((const float4*)A0,
                                                                (float4*)M3, (float4*)M4, NBCC / 4);
    spmm_kernel<<<EE, 384, 0, stream>>>(M1, M3, wf, src, dst, 2.0f);   // x2f = 2*P(x1f) - x0
    spmm_kernel<<<EE, 384, 0, stream>>>(M2, M4, wb, dst, src, 2.0f);   // x2b = 2*P(x1b) - x0

    cheb_gemm_kernel<128, 0><<<NB / 64, 256, 0, stream>>>(A0, M1, M2, M3, M4, ruP, ruB, RU);

    // 2) xrh: keep x-half of A0, overwrite h-half with r*h
    build_xrh_kernel<<<(NB * COUT + TB - 1) / TB, TB, 0, stream>>>(RU, h, A0);

    // ---------- Chebyshev pass 2 (candidate c) ----------
    fill0_kernel<<<(2 * NBCC / 4 + TB - 1) / TB, TB, 0, stream>>>((float4*)M1, 2 * NBCC / 4);
    spmm_kernel<<<EE, 384, 0, stream>>>(A0, M1, wf, src, dst, 1.0f);
    spmm_kernel<<<EE, 384, 0, stream>>>(A0, M2, wb, dst, src, 1.0f);
    initneg_kernel<<<(NBCC / 4 + TB - 1) / TB, TB, 0, stream>>>((const float4*)A0,
                                                                (float4*)M3, (float4*)M4, NBCC / 4);
    spmm_kernel<<<EE, 384, 0, stream>>>(M1, M3, wf, src, dst, 2.0f);
    spmm_kernel<<<EE, 384, 0, stream>>>(M2, M4, wb, dst, src, 2.0f);

    cheb_gemm_kernel<64, 1><<<NB / 64, 128, 0, stream>>>(A0, M1, M2, M3, M4, cP, cB, CBuf);

    // 3) out = u*h + (1-u)*c
    final_kernel<<<(BB * NN * COUT + TB - 1) / TB, TB, 0, stream>>>(RU, CBuf, h, out);

    (void)in_sizes; (void)n_in; (void)out_size; (void)ws_size;
}